// BiLSTM_CRF_27943057228533
// MI455X (gfx1250) — compile-verified
//
#include <hip/hip_runtime.h>

typedef float v2f __attribute__((ext_vector_type(2)));
typedef float v8f __attribute__((ext_vector_type(8)));

#define T        8192
#define EMB      32
#define H2       128
#define G4       512   // 4*H2
#define HID      256
#define KTAG     11
#define START_IDX 9
#define END_IDX  10
#define FK       16    // padded feats row width

// ---------------------------------------------------------------------------
// Phase 2 (listed first so its codegen shows in the disasm snippet):
// sequential LSTM, one workgroup per direction. Thread j owns w_hh row j in
// registers (64 x v2f); h broadcast via LDS as float4 loads; packed f32 FMAs.
// ---------------------------------------------------------------------------
__device__ __forceinline__ float sigmoid_f(float x) {
  return 1.f / (1.f + __expf(-x));
}
__device__ __forceinline__ float tanh_f(float x) {
  float e = __expf(2.f * x);            // inf-safe: 1 - 2/(inf+1) = 1
  return 1.f - 2.f / (e + 1.f);
}

__global__ __launch_bounds__(512) void lstm_kernel(
    const float* __restrict__ prebuf,
    const float* __restrict__ w_hh_f, const float* __restrict__ w_hh_b,
    const float* __restrict__ h0, const float* __restrict__ c0,
    float* __restrict__ hs)
{
  int dir = blockIdx.x;
  int j   = threadIdx.x;
  const float* pre = prebuf + (size_t)dir * T * G4;
  const float* whh = dir ? w_hh_b : w_hh_f;

  // 128 weights in registers as 64 x float2 -> v_pk_fma_f32 operands
  v2f w2[64];
  const v2f* wsrc = (const v2f*)(whh + (size_t)j * H2);
#pragma unroll
  for (int k = 0; k < 64; ++k) w2[k] = wsrc[k];

  __shared__ __align__(16) float h_sh[H2];
  __shared__ float z_sh[G4];

  float c = 0.f;
  if (j < H2) {
    h_sh[j] = h0[dir * H2 + j];
    c       = c0[dir * H2 + j];
  }
  __syncthreads();

  for (int t = 0; t < T; ++t) {
    if (t + 8 < T) __builtin_prefetch(pre + (size_t)(t + 8) * G4 + j, 0, 1);
    float pj = pre[(size_t)t * G4 + j];

    const float4* hv4 = (const float4*)h_sh;   // 32 x ds_load_b128
    v2f a0 = {0.f, 0.f}, a1 = {0.f, 0.f}, a2 = {0.f, 0.f}, a3 = {0.f, 0.f};
#pragma unroll
    for (int q = 0; q < 32; q += 2) {
      float4 ha = hv4[q];
      float4 hb = hv4[q + 1];
      v2f ha0; ha0.x = ha.x; ha0.y = ha.y;
      v2f ha1; ha1.x = ha.z; ha1.y = ha.w;
      v2f hb0; hb0.x = hb.x; hb0.y = hb.y;
      v2f hb1; hb1.x = hb.z; hb1.y = hb.w;
      a0 += w2[2 * q]     * ha0;
      a1 += w2[2 * q + 1] * ha1;
      a2 += w2[2 * q + 2] * hb0;
      a3 += w2[2 * q + 3] * hb1;
    }
    v2f s = (a0 + a1) + (a2 + a3);
    float z = pj + s.x + s.y;

    // each thread applies ITS OWN gate's nonlinearity (parallel across 512)
    float act;
    if (j < 2 * H2)      act = sigmoid_f(z);   // i (0..127), f (128..255)
    else if (j < 3 * H2) act = tanh_f(z);      // g
    else                 act = sigmoid_f(z);   // o
    z_sh[j] = act;
    __syncthreads();                     // activations ready; h_sh reads done

    if (j < H2) {
      float ig = z_sh[j];
      float fg = z_sh[j + H2];
      float gg = z_sh[j + 2 * H2];
      float og = z_sh[j + 3 * H2];
      c = fg * c + ig * gg;
      float h = og * tanh_f(c);
      h_sh[j] = h;
      int ti = dir ? (T - 1 - t) : t;    // un-reverse backward outputs
      hs[(size_t)ti * HID + dir * H2 + j] = h;
    }
    __syncthreads();                     // h_sh updated for next step
  }
}

// ---------------------------------------------------------------------------
// Phase 1: embedding gather + input projection  pre[dir][s][n] (scan order)
// One wave per 16(time) x 16(out) tile, K=32 via 8 chained f32 16x16x4 WMMA.
// ---------------------------------------------------------------------------
__global__ __launch_bounds__(32) void proj_kernel(
    const int* __restrict__ sentence,
    const float* __restrict__ embed_table,
    const float* __restrict__ w_ih_f, const float* __restrict__ b_f,
    const float* __restrict__ w_ih_b, const float* __restrict__ b_b,
    float* __restrict__ prebuf)
{
  int bid  = blockIdx.x;
  int dir  = bid / (512 * 32);          // (T/16) * (G4/16)
  int rem  = bid % (512 * 32);
  int tt   = rem / 32;
  int tn   = rem % 32;
  int t0   = tt * 16;
  int n0   = tn * 16;
  int lane = threadIdx.x;
  int m    = lane & 15;                 // A row (time) / B col (unit)
  int koff = (lane >> 4) * 2;           // K sub-offset per fragment layout

  int tscan = t0 + m;
  int treal = dir ? (T - 1 - tscan) : tscan;
  int word  = sentence[treal];
  const float* erow = embed_table + (size_t)word * EMB;

  const float* wih  = dir ? w_ih_b : w_ih_f;
  const float* wrow = wih + (size_t)(n0 + m) * EMB;   // B col n = n0 + (lane&15)
  const float* bias = dir ? b_b : b_f;

  v8f acc = {0.f, 0.f, 0.f, 0.f, 0.f, 0.f, 0.f, 0.f};
#pragma unroll
  for (int kk = 0; kk < 8; ++kk) {
    int kb = kk * 4 + koff;
    v2f a; a.x = erow[kb]; a.y = erow[kb + 1];
    v2f b; b.x = wrow[kb]; b.y = wrow[kb + 1];
    acc = __builtin_amdgcn_wmma_f32_16x16x4_f32(false, a, false, b,
                                                (short)0, acc, false, false);
  }

  int   n     = n0 + m;
  float bn    = bias[n];
  int   mbase = (lane >> 4) * 8;        // C/D layout: VGPR r -> M=r (+8 upper)
#pragma unroll
  for (int r = 0; r < 8; ++r) {
    int mm = mbase + r;
    prebuf[((size_t)dir * T + (size_t)(t0 + mm)) * G4 + n] = acc[r] + bn;
  }
}

// ---------------------------------------------------------------------------
// Phase 3: feats = hs @ W_out^T + b_out, N padded 11->16, K=256 via WMMA f32.
// ---------------------------------------------------------------------------
__global__ __launch_bounds__(32) void feats_kernel(
    const float* __restrict__ hs,
    const float* __restrict__ W_out, const float* __restrict__ b_out,
    float* __restrict__ feats)
{
  int t0   = blockIdx.x * 16;
  int lane = threadIdx.x;
  int m    = lane & 15;
  int koff = (lane >> 4) * 2;

  const float* arow  = hs + (size_t)(t0 + m) * HID;
  bool  valid = (m < KTAG);
  float scale = valid ? 1.f : 0.f;       // zero B columns 11..15
  const float* wrow = W_out + (size_t)(valid ? m : 0) * HID;

  v8f acc = {0.f, 0.f, 0.f, 0.f, 0.f, 0.f, 0.f, 0.f};
#pragma unroll
  for (int kk = 0; kk < 64; ++kk) {
    int kb = kk * 4 + koff;
    v2f a; a.x = arow[kb];          a.y = arow[kb + 1];
    v2f b; b.x = wrow[kb] * scale;  b.y = wrow[kb + 1] * scale;
    acc = __builtin_amdgcn_wmma_f32_16x16x4_f32(false, a, false, b,
                                                (short)0, acc, false, false);
  }

  int mbase = (lane >> 4) * 8;
  if (valid) {
    float bn = b_out[m];
#pragma unroll
    for (int r = 0; r < 8; ++r) {
      int mm = mbase + r;
      feats[(size_t)(t0 + mm) * FK + m] = acc[r] + bn;
    }
  }
}

// ---------------------------------------------------------------------------
// Phase 4: Viterbi forward scan + backtrace, single wave. feats and
// backpointers staged chunk-wise through LDS to keep the dependent chains
// at LDS latency.
// ---------------------------------------------------------------------------
#define VCH 512

__global__ __launch_bounds__(32) void viterbi_kernel(
    const float* __restrict__ feats,
    const float* __restrict__ trans,
    int* __restrict__ bp,
    float* __restrict__ out)
{
  int lane = threadIdx.x;
  __shared__ __align__(16) float fsh[VCH * FK];
  __shared__ int   bpsh[VCH * KTAG];
  __shared__ float tsh[32];
  __shared__ int   tagsh;

  float tr[KTAG];
#pragma unroll
  for (int p = 0; p < KTAG; ++p)
    tr[p] = (lane < KTAG) ? trans[lane * KTAG + p] : 0.f;

  float fv = (lane == START_IDX) ? 0.f : -10000.f;

  // ---- forward scan ----
  for (int chunk = 0; chunk < T / VCH; ++chunk) {
    int base = chunk * VCH;
    const float4* src = (const float4*)(feats + (size_t)base * FK);
    float4*       dst = (float4*)fsh;
    for (int i = lane; i < VCH * FK / 4; i += 32) dst[i] = src[i];
    __syncthreads();

    for (int tt = 0; tt < VCH; ++tt) {
      float best = -3.4e38f;
      int   bidx = 0;
#pragma unroll
      for (int p = 0; p < KTAG; ++p) {
        float fvp = __shfl(fv, p, 32);
        float sc  = fvp + tr[p];
        if (sc > best) { best = sc; bidx = p; }   // first-max tie break
      }
      float nf = best + fsh[tt * FK + lane];
      if (lane < KTAG) {
        bp[(size_t)(base + tt) * KTAG + lane] = bidx;
        fv = nf;
      }
    }
    __syncthreads();
  }

  // ---- terminal + argmax ----
  float term = fv + ((lane < KTAG) ? trans[END_IDX * KTAG + lane] : 0.f);
  tsh[lane] = (lane < KTAG) ? term : -3.4e38f;
  __syncthreads();
  if (lane == 0) {
    float bestv = tsh[0];
    int   bi = 0;
    for (int p = 1; p < KTAG; ++p)
      if (tsh[p] > bestv) { bestv = tsh[p]; bi = p; }
    out[0] = bestv;
    tagsh  = bi;
  }
  __syncthreads();

  // ---- backtrace (lane 0 walks LDS-staged chunks) ----
  int tag = tagsh;
  for (int chunk = T / VCH - 1; chunk >= 0; --chunk) {
    int base = chunk * VCH;
    for (int i = lane; i < VCH * KTAG; i += 32)
      bpsh[i] = bp[(size_t)base * KTAG + i];
    __syncthreads();
    if (lane == 0) {
      for (int tt = VCH - 1; tt >= 0; --tt) {
        out[1 + base + tt] = (float)tag;
        tag = bpsh[tt * KTAG + tag];
      }
    }
    __syncthreads();
  }
}

// ---------------------------------------------------------------------------
extern "C" void kernel_launch(void* const* d_in, const int* in_sizes, int n_in,
                              void* d_out, int out_size, void* d_ws, size_t ws_size,
                              hipStream_t stream) {
  const int*   sentence = (const int*)d_in[0];
  const float* embed    = (const float*)d_in[1];
  const float* w_ih_f   = (const float*)d_in[2];
  const float* w_hh_f   = (const float*)d_in[3];
  const float* b_f      = (const float*)d_in[4];
  const float* w_ih_b   = (const float*)d_in[5];
  const float* w_hh_b   = (const float*)d_in[6];
  const float* b_b      = (const float*)d_in[7];
  const float* W_out    = (const float*)d_in[8];
  const float* b_out    = (const float*)d_in[9];
  const float* trans    = (const float*)d_in[10];
  const float* h0       = (const float*)d_in[11];
  const float* c0       = (const float*)d_in[12];

  char*  ws     = (char*)d_ws;
  size_t off    = 0;
  float* prebuf = (float*)(ws + off); off += (size_t)2 * T * G4 * sizeof(float);   // 32 MB
  float* hs     = (float*)(ws + off); off += (size_t)T * HID * sizeof(float);      //  8 MB
  float* feats  = (float*)(ws + off); off += (size_t)T * FK * sizeof(float);       // .5 MB
  int*   bp     = (int*)  (ws + off); off += (size_t)T * KTAG * sizeof(int);       // .36 MB

  proj_kernel <<<2 * 512 * 32, 32, 0, stream>>>(sentence, embed,
                                                w_ih_f, b_f, w_ih_b, b_b, prebuf);
  lstm_kernel <<<2, 512, 0, stream>>>(prebuf, w_hh_f, w_hh_b, h0, c0, hs);
  feats_kernel<<<512, 32, 0, stream>>>(hs, W_out, b_out, feats);
  viterbi_kernel<<<1, 32, 0, stream>>>(feats, trans, bp, (float*)d_out);
}